// StandardAUTKCLoss_30081950941417
// MI455X (gfx1250) — compile-verified
//
#include <hip/hip_runtime.h>
#include <hip/hip_bf16.h>
#include <stdint.h>

#define THREADS      256
#define NCLASS       32000
#define CHUNK_FLOATS 8000
#define CHUNK_F4     2000
#define NCHUNK       4
#define TRIPS        8
#define TOPN         7
#define NEG_INF      (-__builtin_inff())

// Issue one 8000-float chunk as per-lane async b128 copies global -> LDS.
// Every thread issues exactly TRIPS instructions (tail clamped to a duplicate
// address writing identical data), so ASYNCcnt per wave per chunk == 8.
__device__ __forceinline__ void issue_chunk_async(const float* gbase,
                                                  unsigned lds_base, int tid) {
#pragma unroll
  for (int k = 0; k < TRIPS; ++k) {
    int i = tid + (k << 8);
    if (i > CHUNK_F4 - 1) i = CHUNK_F4 - 1;
    unsigned lds_addr = lds_base + ((unsigned)i << 4);
    unsigned goff     = (unsigned)i << 4;
    asm volatile("global_load_async_to_lds_b128 %0, %1, %2"
                 :: "v"(lds_addr), "v"(goff), "s"(gbase)
                 : "memory");
  }
}

// Branchless insert of x into descending sorted t[0..6] (register-resident,
// all indices compile-time after unroll -> pure v_cndmask chains).
__device__ __forceinline__ void topk_insert(float t[TOPN], float x) {
  float n0 = (x > t[0]) ? x : t[0];
#pragma unroll
  for (int k = TOPN - 1; k >= 1; --k)
    t[k] = (x > t[k]) ? ((x > t[k - 1]) ? t[k - 1] : x) : t[k];
  t[0] = n0;
}

// Merge two descending sorted 7-lists into top-7 (shift-register style,
// fully unrolled, no dynamic register indexing).
__device__ __forceinline__ void merge7(float a[TOPN], const float bin[TOPN]) {
  float b[TOPN], r[TOPN];
#pragma unroll
  for (int k = 0; k < TOPN; ++k) b[k] = bin[k];
#pragma unroll
  for (int k = 0; k < TOPN; ++k) {
    bool ta = a[0] >= b[0];
    r[k] = ta ? a[0] : b[0];
#pragma unroll
    for (int q = 0; q < TOPN - 1; ++q) {
      a[q] = ta ? a[q + 1] : a[q];
      b[q] = ta ? b[q] : b[q + 1];
    }
    a[TOPN - 1] = ta ? NEG_INF : a[TOPN - 1];
    b[TOPN - 1] = ta ? b[TOPN - 1] : NEG_INF;
  }
#pragma unroll
  for (int k = 0; k < TOPN; ++k) a[k] = r[k];
}

__global__ void __launch_bounds__(THREADS)
autkc_row_kernel(const float* __restrict__ pred, const int* __restrict__ y,
                 float* __restrict__ row_loss) {
  __shared__ float4 sbuf[2][CHUNK_F4];   // 64000 B double buffer
  __shared__ float  red[8][12];          // per-wave: t[0..6], m, s
  __shared__ float  s_xpos;

  const int row  = blockIdx.x;
  const int tid  = threadIdx.x;
  const int wave = tid >> 5;
  const int lane = tid & 31;
  const float* rowp = pred + (size_t)row * NCLASS;

  if (tid == 0) s_xpos = rowp[y[row]];   // positive-class logit

  const unsigned buf0 = (unsigned)(uintptr_t)(&sbuf[0][0]);
  const unsigned buf1 = (unsigned)(uintptr_t)(&sbuf[1][0]);

  issue_chunk_async(rowp, buf0, tid);
  issue_chunk_async(rowp + CHUNK_FLOATS, buf1, tid);

  float t[TOPN];
#pragma unroll
  for (int k = 0; k < TOPN; ++k) t[k] = NEG_INF;
  float m = NEG_INF;
  float s0 = 0.f, s1 = 0.f, s2 = 0.f, s3 = 0.f;

  for (int c = 0; c < NCHUNK; ++c) {
    if (c < NCHUNK - 1) asm volatile("s_wait_asynccnt 0x8" ::: "memory");
    else                asm volatile("s_wait_asynccnt 0x0" ::: "memory");
    __syncthreads();
    const float4* buf = sbuf[c & 1];
#pragma unroll
    for (int k = 0; k < TRIPS; ++k) {
      int i = tid + (k << 8);
      if (i < CHUNK_F4) {
        float4 v = buf[i];
        float mx = fmaxf(fmaxf(v.x, v.y), fmaxf(v.z, v.w));
        if (mx > m) {                      // rare: rescale online softmax
          float r = __expf(m - mx);
          s0 *= r; s1 *= r; s2 *= r; s3 *= r;
          m = mx;
        }
        s0 += __expf(v.x - m);
        s1 += __expf(v.y - m);
        s2 += __expf(v.z - m);
        s3 += __expf(v.w - m);
        if (mx > t[TOPN - 1]) {            // rare: touch the top-7 list
          topk_insert(t, v.x); topk_insert(t, v.y);
          topk_insert(t, v.z); topk_insert(t, v.w);
        }
      }
    }
    __syncthreads();                       // everyone done reading buf[c&1]
    if (c + 2 < NCHUNK)
      issue_chunk_async(rowp + (size_t)(c + 2) * CHUNK_FLOATS,
                        (c & 1) ? buf1 : buf0, tid);
  }

  float s = (s0 + s1) + (s2 + s3);

  // Wave32 reduction via shuffles (ds-permute path).
#pragma unroll
  for (int off = 16; off >= 1; off >>= 1) {
    float o[TOPN];
#pragma unroll
    for (int k = 0; k < TOPN; ++k) o[k] = __shfl_xor(t[k], off, 32);
    merge7(t, o);
    float mo = __shfl_xor(m, off, 32);
    float so = __shfl_xor(s, off, 32);
    float mn = fmaxf(m, mo);
    s = s * __expf(m - mn) + so * __expf(mo - mn);
    m = mn;
  }

  if (lane == 0) {
#pragma unroll
    for (int k = 0; k < TOPN; ++k) red[wave][k] = t[k];
    red[wave][7] = m;
    red[wave][8] = s;
  }
  __syncthreads();

  if (tid == 0) {
    float T[TOPN];
#pragma unroll
    for (int k = 0; k < TOPN; ++k) T[k] = red[0][k];
    float M = red[0][7], S = red[0][8];
    for (int w = 1; w < 8; ++w) {
      float o[TOPN];
#pragma unroll
      for (int k = 0; k < TOPN; ++k) o[k] = red[w][k];
      merge7(T, o);
      float mo = red[w][7], so = red[w][8];
      float mn = fmaxf(M, mo);
      S = S * __expf(M - mn) + so * __expf(mo - mn);
      M = mn;
    }
    // Drop the positive logit from the merged top-7 -> top-6 negatives.
    float xp = s_xpos;
    float out6[6];
    bool rm = false;
#pragma unroll
    for (int k = 0; k < 6; ++k) {
      rm = rm || (T[k] == xp);
      out6[k] = rm ? T[k + 1] : T[k];
    }
    float inv_s = 1.0f / S;
    float ppos  = __expf(xp - M) * inv_s;
    float acc = 0.f;
#pragma unroll
    for (int k = 0; k < 6; ++k) {
      float p = __expf(out6[k] - M) * inv_s;
      float d = 1.0f + p - ppos;
      acc += d * d;
    }
    row_loss[row] = acc * 0.2f;            // /K, K=5
  }
}

// Deterministic fixed-order mean over B row losses.
__global__ void __launch_bounds__(256)
mean_kernel(const float* __restrict__ v, float* __restrict__ out, int n) {
  __shared__ float sm[256];
  int tid = threadIdx.x;
  float a = 0.f;
  for (int i = tid; i < n; i += 256) a += v[i];
  sm[tid] = a;
  __syncthreads();
  for (int w = 128; w > 0; w >>= 1) {
    if (tid < w) sm[tid] += sm[tid + w];
    __syncthreads();
  }
  if (tid == 0) out[0] = sm[0] / (float)n;
}

extern "C" void kernel_launch(void* const* d_in, const int* in_sizes, int n_in,
                              void* d_out, int out_size, void* d_ws, size_t ws_size,
                              hipStream_t stream) {
  const float* pred = (const float*)d_in[0];
  const int*   y    = (const int*)d_in[1];   // JAX default: int64 demoted to int32
  float* out = (float*)d_out;
  float* ws  = (float*)d_ws;                 // B floats of per-row loss
  const int B = in_sizes[1];

  autkc_row_kernel<<<B, THREADS, 0, stream>>>(pred, y, ws);
  mean_kernel<<<1, 256, 0, stream>>>(ws, out, B);
}